// LargeEmbeddings_18021682774354
// MI455X (gfx1250) — compile-verified
//
#include <hip/hip_runtime.h>

// ---------------------------------------------------------------------------
// EmbeddingBag sum-pool for MI455X (gfx1250).
// 4 tables [500000,128] fp32, 81920 ids/segs per table, output [4096, 512].
// Memory-bound (~180 MB -> ~8us @ 23.3 TB/s): no matmul, so no WMMA.
// Uses the CDNA5 Tensor Data Mover in GATHER mode (32-bit row indices,
// 8 rows / 4KB per DMA) to pull embedding rows into LDS, double-buffered
// via TENSORcnt. All per-wave-uniform control (binary search, descriptor
// build, gather indices) is forced onto the scalar pipe via readfirstlane.
// ---------------------------------------------------------------------------

#define NUM_EMB        500000
#define DIM            128
#define BATCH          4096
#define N_TABLES       4
#define ROWS_PER_DMA   8          // 32-bit gather indices: up to 8 rows per D#
#define WAVES_PER_BLOCK 4
#define BLOCK_THREADS  (WAVES_PER_BLOCK * 32)

typedef __attribute__((ext_vector_type(4))) unsigned int u32x4;
typedef __attribute__((ext_vector_type(8))) int          i32x8;
typedef __attribute__((ext_vector_type(4))) int          i32x4;

#if __has_builtin(__builtin_amdgcn_tensor_load_to_lds) && __has_builtin(__builtin_amdgcn_s_wait_tensorcnt)
#define HAS_TDM 1
#else
#define HAS_TDM 0
#endif

// first index i in [0,n) with seg[i] >= key (seg sorted ascending).
// All inputs are wave-uniform scalars -> SALU loop with s_load.
__device__ __forceinline__ int lower_bound_seg(const int* __restrict__ seg, int n, int key) {
  int lo = 0, hi = n;
  while (lo < hi) {
    int mid = (lo + hi) >> 1;
    if (seg[mid] < key) lo = mid + 1; else hi = mid;
  }
  return lo;
}

#if HAS_TDM
// Issue one TDM gather: rows ids[first .. first+n) of `tab` -> the LDS address in lds_byte_addr,
// packed contiguously (n * 128 floats). n in [1, 8]. All args wave-uniform.
__device__ __forceinline__ void tdm_gather_rows(unsigned long long tab_addr,
                                                const int* __restrict__ ids,
                                                int first, int n,
                                                unsigned lds_byte_addr) {
  // ---- D# group 0 (4 SGPR dwords) ----
  u32x4 g0;
  g0[0] = 1u               // count = 1 (valid descriptor)
        | (1u << 30)       // gather_index_size = 32-bit
        | (1u << 31);      // gather_mode = 1
  g0[1] = lds_byte_addr;   // LDS destination (bytes)
  g0[2] = (unsigned)(tab_addr & 0xffffffffull);              // global_addr[31:0]
  g0[3] = (unsigned)((tab_addr >> 32) & 0x1ffffffull)        // global_addr[56:32]
        | (2u << 30);                                        // type = 2 ("image")

  // ---- D# group 1 (8 SGPR dwords) ----
  i32x8 g1;
  g1[0] = (2 << 16);                                         // data_size = 4 bytes; wg_mask=0
  g1[1] = (int)((unsigned)DIM << 16);                        // tensor_dim0[15:0] = 128 (row len)
  g1[2] = (int)(((unsigned)NUM_EMB & 0xffffu) << 16);        // tensor_dim1[15:0] (num rows)
  g1[3] = (int)(((unsigned)NUM_EMB >> 16)                    // tensor_dim1[31:16]
        | ((unsigned)DIM << 16));                            // tile_dim0 = 128
  g1[4] = n;                                                 // tile_dim1 = #valid indices
  g1[5] = DIM;                                               // tensor_dim0_stride[31:0] = 128
  g1[6] = 0;                                                 // stride hi / dim1_stride (ignored)
  g1[7] = 0;

  // ---- D# groups 2/3: 32-bit row indices (branchless, clamped in-bounds;
  //      TDM ignores indices >= tile_dim1) ----
  const int last = first + n - 1;
  int idx[8];
#pragma unroll
  for (int j = 0; j < 8; ++j) {
    int q = first + j;
    q = (q < last) ? q : last;
    idx[j] = ids[q];
  }
  i32x4 g2 = {idx[0], idx[1], idx[2], idx[3]};
  i32x4 g3 = {idx[4], idx[5], idx[6], idx[7]};

#if defined(__clang_major__) && __clang_major__ >= 23
  i32x8 g1b = {};  // amdgpu-toolchain (clang-23 / therock-10.0) 6-arg form
  __builtin_amdgcn_tensor_load_to_lds(g0, g1, g2, g3, g1b, 0);
#else
  __builtin_amdgcn_tensor_load_to_lds(g0, g1, g2, g3, 0);     // ROCm 7.2 5-arg form
#endif
}
#endif  // HAS_TDM

__global__ __launch_bounds__(BLOCK_THREADS) void embbag_pool_kernel(
    const float* __restrict__ t0, const int* __restrict__ i0, const int* __restrict__ s0,
    const float* __restrict__ t1, const int* __restrict__ i1, const int* __restrict__ s1,
    const float* __restrict__ t2, const int* __restrict__ i2, const int* __restrict__ s2,
    const float* __restrict__ t3, const int* __restrict__ i3, const int* __restrict__ s3,
    float* __restrict__ out, int total) {
#if HAS_TDM
  __shared__ float smem[WAVES_PER_BLOCK][2][ROWS_PER_DMA * DIM];  // 32 KB
#endif
  const int lane = (int)(threadIdx.x & 31u);
  // Force wave-uniform values onto the scalar pipe.
  const int wave = __builtin_amdgcn_readfirstlane((int)(threadIdx.x >> 5u));
  const int gw   = (int)blockIdx.x * WAVES_PER_BLOCK + wave;   // 0 .. 16383
  const int table = __builtin_amdgcn_readfirstlane(gw >> 12);  // / BATCH
  const int ex    = __builtin_amdgcn_readfirstlane(gw & (BATCH - 1));

  const float* tab; const int* ids; const int* seg;
  if      (table == 0) { tab = t0; ids = i0; seg = s0; }
  else if (table == 1) { tab = t1; ids = i1; seg = s1; }
  else if (table == 2) { tab = t2; ids = i2; seg = s2; }
  else                 { tab = t3; ids = i3; seg = s3; }

  // seg is sorted: this wave owns ids[lo..hi) (wave-uniform binary search)
  const int lo = __builtin_amdgcn_readfirstlane(lower_bound_seg(seg, total, ex));
  const int hi = __builtin_amdgcn_readfirstlane(lower_bound_seg(seg, total, ex + 1));

  float4 acc = make_float4(0.f, 0.f, 0.f, 0.f);

#if HAS_TDM
  const unsigned long long tab_addr = (unsigned long long)(size_t)tab;
  // LDS byte offsets of the two per-wave buffers (low 32 bits of generic addr)
  const unsigned ldsA = (unsigned)(size_t)(&smem[wave][0][0]);
  const unsigned ldsB = (unsigned)(size_t)(&smem[wave][1][0]);

  int i = lo, buf = 0;
  if (i < hi) {
    int n0 = hi - i; if (n0 > ROWS_PER_DMA) n0 = ROWS_PER_DMA;
    tdm_gather_rows(tab_addr, ids, i, n0, ldsA);
  }
  while (i < hi) {
    int n = hi - i; if (n > ROWS_PER_DMA) n = ROWS_PER_DMA;
    const int ni = i + n;
    if (ni < hi) {
      int n2 = hi - ni; if (n2 > ROWS_PER_DMA) n2 = ROWS_PER_DMA;
      tdm_gather_rows(tab_addr, ids, ni, n2, buf ? ldsA : ldsB);  // prefetch next chunk
      __builtin_amdgcn_s_wait_tensorcnt(1);   // current chunk's DMA complete
    } else {
      __builtin_amdgcn_s_wait_tensorcnt(0);   // last chunk: drain
    }
    __asm__ volatile("" ::: "memory");        // fence LDS reads below the wait

    const float* bp = &smem[wave][buf][0];
#pragma unroll 4
    for (int r = 0; r < n; ++r) {
      const float4 v = *(const float4*)(bp + r * DIM + lane * 4);  // ds_load_b128
      acc.x += v.x; acc.y += v.y; acc.z += v.z; acc.w += v.w;
    }
    i = ni; buf ^= 1;
  }
#else
  // Fallback: direct vectorized gather, 4-deep unrolled for memory-level parallelism
  int i = lo;
  for (; i + 4 <= hi; i += 4) {
    const int r0 = ids[i], r1 = ids[i + 1], r2 = ids[i + 2], r3 = ids[i + 3];
    const float4 v0 = *(const float4*)(tab + (size_t)r0 * DIM + lane * 4);
    const float4 v1 = *(const float4*)(tab + (size_t)r1 * DIM + lane * 4);
    const float4 v2 = *(const float4*)(tab + (size_t)r2 * DIM + lane * 4);
    const float4 v3 = *(const float4*)(tab + (size_t)r3 * DIM + lane * 4);
    acc.x += v0.x + v1.x + v2.x + v3.x;
    acc.y += v0.y + v1.y + v2.y + v3.y;
    acc.z += v0.z + v1.z + v2.z + v3.z;
    acc.w += v0.w + v1.w + v2.w + v3.w;
  }
  for (; i < hi; ++i) {
    const int r = ids[i];
    const float4 v = *(const float4*)(tab + (size_t)r * DIM + lane * 4);
    acc.x += v.x; acc.y += v.y; acc.z += v.z; acc.w += v.w;
  }
#endif

  // out[ex, table*128 + lane*4 .. +4]  (KeyedTensor concat along features)
  float4* op = (float4*)(out + (size_t)ex * (N_TABLES * DIM) + table * DIM + lane * 4);
  *op = acc;
}

extern "C" void kernel_launch(void* const* d_in, const int* in_sizes, int n_in,
                              void* d_out, int out_size, void* d_ws, size_t ws_size,
                              hipStream_t stream) {
  (void)n_in; (void)out_size; (void)d_ws; (void)ws_size;
  // setup_inputs order: table0, ids0, seg0, table1, ids1, seg1, ...
  const float* t0 = (const float*)d_in[0];
  const int*   i0 = (const int*)  d_in[1];
  const int*   s0 = (const int*)  d_in[2];
  const float* t1 = (const float*)d_in[3];
  const int*   i1 = (const int*)  d_in[4];
  const int*   s1 = (const int*)  d_in[5];
  const float* t2 = (const float*)d_in[6];
  const int*   i2 = (const int*)  d_in[7];
  const int*   s2 = (const int*)  d_in[8];
  const float* t3 = (const float*)d_in[9];
  const int*   i3 = (const int*)  d_in[10];
  const int*   s3 = (const int*)  d_in[11];
  const int total = in_sizes[1];  // 81920

  const int total_waves = N_TABLES * BATCH;                 // 16384
  const int blocks = total_waves / WAVES_PER_BLOCK;         // 4096
  embbag_pool_kernel<<<blocks, BLOCK_THREADS, 0, stream>>>(
      t0, i0, s0, t1, i1, s1, t2, i2, s2, t3, i3, s3, (float*)d_out, total);
}